// Advanced_RGCN_3367254360423
// MI455X (gfx1250) — compile-verified
//
#include <hip/hip_runtime.h>
#include <math.h>

typedef __attribute__((ext_vector_type(2))) float v2f;
typedef __attribute__((ext_vector_type(8))) float v8f;

#define N_NODES   50000
#define NUM_REL   8
#define IN_DIM    128
#define HID       256
#define NUM_EDGES 1600000
#define M_DEC     200000
#define LN_EPS    1e-5f

// ---------------------------------------------------------------------------
// Edge aggregation: agg[dst, r, :] += x[src, :]; cnt[dst, r] += 1
// One thread per (edge, channel). d is a power of two (128 or 256).
// ---------------------------------------------------------------------------
__global__ void rgcn_agg_kernel(const float* __restrict__ x,
                                const int* __restrict__ edge_index,
                                const int* __restrict__ edge_type,
                                float* __restrict__ agg,
                                float* __restrict__ cnt,
                                int E, int d, int dshift) {
  long idx = (long)blockIdx.x * blockDim.x + threadIdx.x;
  long total = (long)E << dshift;
  if (idx >= total) return;
  int e = (int)(idx >> dshift);
  int c = (int)(idx & (d - 1));
  int src = edge_index[e];        // edge_index[0, e]
  int dst = edge_index[E + e];    // edge_index[1, e]
  int r   = edge_type[e];
  atomicAdd(&agg[((size_t)dst * NUM_REL + r) * d + c], x[(size_t)src * d + c]);
  if (c == 0) atomicAdd(&cnt[dst * NUM_REL + r], 1.0f);
}

__global__ void rcnt_kernel(float* cnt, int n) {
  int i = blockIdx.x * blockDim.x + threadIdx.x;
  if (i < n) cnt[i] = 1.0f / fmaxf(cnt[i], 1.0f);
}

// ---------------------------------------------------------------------------
// fp32 WMMA GEMM: C[M,N] = act( (ACCUM? C:0) + A[M,K] @ B[K,N] + bias )
// Block: 128 threads (4 waves, 2x2), 64x128 block tile, wave tile 32x64,
// K-step 16, double-buffered LDS software pipeline.
// LDS tiles are pair-interleaved along K so every WMMA fragment is one
// naturally aligned ds_load_b64 into the operand VGPR pair.
// AMODE 0: plain A (lda = K)
// AMODE 1: A element scaled by a_scale[row*NUM_REL + (k >> sshift)] (mean agg)
// AMODE 2: gathered concat rows: A row m = concat(h[dec[2m]], h[dec[2m+1]]),
//          h width HID (=256), K = 2*HID = 512
// ACT 0: none, 1: exact-erf GELU
// Requires: N % 128 == 0 or N == 128-tile grid, K % 16 == 0.
// ---------------------------------------------------------------------------
__device__ __forceinline__ float gelu_exact(float v) {
  return 0.5f * v * (1.0f + erff(v * 0.70710678118654752f));
}

template <int AMODE, int ACT, int BIAS, int ACCUM>
__global__ __launch_bounds__(128)
void gemm_wmma_kernel(float* __restrict__ C,
                      const float* __restrict__ A,
                      const float* __restrict__ B,
                      const float* __restrict__ bias,
                      const float* __restrict__ a_scale,
                      const int* __restrict__ dec,
                      int M, int N, int K, int sshift) {
  // pair-interleaved: element (k, m) lives at [k>>1][m*2 + (k&1)]
  __shared__ float As[2][8][132];   // [buf][kpair][64*2 + pad]
  __shared__ float Bs[2][8][260];   // [buf][kpair][128*2 + pad]

  const int tid  = threadIdx.x;
  const int lane = tid & 31;
  const int half = lane >> 4;
  const int l16  = lane & 15;
  const int wave = tid >> 5;
  const int wm   = (wave >> 1) * 32;  // wave M-origin in block tile
  const int wn   = (wave & 1) * 64;   // wave N-origin in block tile
  const int m0   = blockIdx.x * 64;
  const int n0   = blockIdx.y * 128;

  float aReg[8];   // A stage: i -> tile row i*8 + (tid>>4), col tid&15
  float bReg[16];  // B stage: i -> tile row i,             col tid

  auto loadA = [&](int k0) {
#pragma unroll
    for (int i = 0; i < 8; i++) {
      int row = m0 + i * 8 + (tid >> 4);
      int k = k0 + (tid & 15);
      float v = 0.0f;
      if (row < M) {
        if (AMODE == 2) {
          int nid = dec[row * 2 + (k >> 8)];          // k<512 -> {0,1}
          v = A[(size_t)nid * HID + (k & (HID - 1))];
        } else {
          v = A[(size_t)row * K + k];
          if (AMODE == 1) v *= a_scale[row * NUM_REL + (k >> sshift)];
        }
      }
      aReg[i] = v;
    }
  };
  auto storeA = [&](int buf) {
#pragma unroll
    for (int i = 0; i < 8; i++) {
      int r = i * 8 + (tid >> 4);
      int c = tid & 15;
      As[buf][c >> 1][r * 2 + (c & 1)] = aReg[i];
    }
  };
  auto loadB = [&](int k0) {
#pragma unroll
    for (int i = 0; i < 16; i++)
      bReg[i] = B[(size_t)(k0 + i) * N + n0 + tid];
  };
  auto storeB = [&](int buf) {
#pragma unroll
    for (int i = 0; i < 16; i++)
      Bs[buf][i >> 1][tid * 2 + (i & 1)] = bReg[i];
  };

  v8f acc[2][4] = {};

  loadA(0);
  loadB(0);
  storeA(0);
  storeB(0);
  __syncthreads();

  int buf = 0;
  for (int k0 = 0; k0 < K; k0 += 16) {
    const int nxt = k0 + 16;
    if (nxt < K) {  // stage next tile into registers (overlaps compute)
      loadA(nxt);
      loadB(nxt);
    }
    if (k0 + 32 < K) {  // prefetch B two K-steps ahead (64B per thread)
      __builtin_prefetch(
          &B[(size_t)(k0 + 32 + (tid >> 3)) * N + n0 + (tid & 7) * 16], 0, 1);
    }

#pragma unroll
    for (int kq = 0; kq < 4; kq++) {
      const int p = kq * 2 + half;  // lanes 0-15: K{4kq,4kq+1}; 16-31: +2
      v2f a[2], b[4];
#pragma unroll
      for (int t = 0; t < 2; t++)
        a[t] = *(const v2f*)&As[buf][p][(wm + t * 16 + l16) * 2];
#pragma unroll
      for (int u = 0; u < 4; u++)
        b[u] = *(const v2f*)&Bs[buf][p][(wn + u * 16 + l16) * 2];
#pragma unroll
      for (int t = 0; t < 2; t++)
#pragma unroll
        for (int u = 0; u < 4; u++)
          acc[t][u] = __builtin_amdgcn_wmma_f32_16x16x4_f32(
              false, a[t], false, b[u], (short)0, acc[t][u], false, false);
    }

    if (nxt < K) {
      storeA(buf ^ 1);
      storeB(buf ^ 1);
    }
    __syncthreads();
    buf ^= 1;
  }

  // ---- epilogue: C/D layout is VGPR v -> row (v + 8*half), col l16 ----
#pragma unroll
  for (int tm = 0; tm < 2; tm++) {
#pragma unroll
    for (int tn = 0; tn < 4; tn++) {
      int col = n0 + wn + tn * 16 + l16;
      float bv = BIAS ? bias[col] : 0.0f;
#pragma unroll
      for (int v = 0; v < 8; v++) {
        int row = m0 + wm + tm * 16 + half * 8 + v;
        if (row < M) {
          float val = acc[tm][tn][v];
          if (ACCUM) val += C[(size_t)row * N + col];
          val += bv;
          if (ACT == 1) val = gelu_exact(val);
          C[(size_t)row * N + col] = val;
        }
      }
    }
  }
}

// ---------------------------------------------------------------------------
// Row-wise relu + LayerNorm over HID=256. One wave per row, 8 rows per block.
// ADDINTO: out[row] += LN(relu(in[row]))  (used to form h = h1 + h2)
// ---------------------------------------------------------------------------
template <bool ADDINTO>
__global__ void relu_ln_kernel(const float* __restrict__ in,
                               const float* __restrict__ g,
                               const float* __restrict__ b,
                               float* __restrict__ out, int rows) {
  int wave = threadIdx.x >> 5;
  int lane = threadIdx.x & 31;
  int row = blockIdx.x * 8 + wave;
  if (row >= rows) return;
  const float* ir = in + (size_t)row * HID;
  float v[8];
  float s = 0.0f;
#pragma unroll
  for (int j = 0; j < 8; j++) {
    v[j] = fmaxf(ir[lane + j * 32], 0.0f);
    s += v[j];
  }
#pragma unroll
  for (int o = 16; o > 0; o >>= 1) s += __shfl_xor(s, o, 32);
  float mu = s * (1.0f / HID);
  float q = 0.0f;
#pragma unroll
  for (int j = 0; j < 8; j++) {
    float d = v[j] - mu;
    q += d * d;
  }
#pragma unroll
  for (int o = 16; o > 0; o >>= 1) q += __shfl_xor(q, o, 32);
  float rs = rsqrtf(q * (1.0f / HID) + LN_EPS);
  float* orow = out + (size_t)row * HID;
#pragma unroll
  for (int j = 0; j < 8; j++) {
    int c = lane + j * 32;
    float val = (v[j] - mu) * rs * g[c] + b[c];
    if (ADDINTO)
      orow[c] += val;
    else
      orow[c] = val;
  }
}

// ---------------------------------------------------------------------------
// Final 128 -> 2 FC (tiny): out[m, :] = z2[m, :] @ w3 + b3
// ---------------------------------------------------------------------------
__global__ void final_fc_kernel(const float* __restrict__ z2,
                                const float* __restrict__ w3,
                                const float* __restrict__ b3,
                                float* __restrict__ out, int M) {
  int m = blockIdx.x * blockDim.x + threadIdx.x;
  if (m >= M) return;
  const float* zr = z2 + (size_t)m * 128;
  float a0 = b3[0], a1 = b3[1];
#pragma unroll 4
  for (int k = 0; k < 128; k++) {
    float v = zr[k];
    a0 += v * w3[k * 2 + 0];
    a1 += v * w3[k * 2 + 1];
  }
  out[m * 2 + 0] = a0;
  out[m * 2 + 1] = a1;
}

// ---------------------------------------------------------------------------
// Launch
// ---------------------------------------------------------------------------
extern "C" void kernel_launch(void* const* d_in, const int* in_sizes, int n_in,
                              void* d_out, int out_size, void* d_ws, size_t ws_size,
                              hipStream_t stream) {
  const float* x      = (const float*)d_in[0];
  const int* ei       = (const int*)d_in[1];
  const int* et       = (const int*)d_in[2];
  const int* dec      = (const int*)d_in[3];
  const float* W1     = (const float*)d_in[4];   // [8,128,256] == [1024,256]
  const float* root1  = (const float*)d_in[5];
  const float* b1     = (const float*)d_in[6];
  const float* W2     = (const float*)d_in[7];   // [8,256,256] == [2048,256]
  const float* root2  = (const float*)d_in[8];
  const float* b2     = (const float*)d_in[9];
  const float* ln1_g  = (const float*)d_in[10];
  const float* ln1_b  = (const float*)d_in[11];
  const float* ln2_g  = (const float*)d_in[12];
  const float* ln2_b  = (const float*)d_in[13];
  const float* mlp_w1 = (const float*)d_in[14];
  const float* mlp_b1 = (const float*)d_in[15];
  const float* mlp_w2 = (const float*)d_in[16];
  const float* mlp_b2 = (const float*)d_in[17];
  const float* mlp_w3 = (const float*)d_in[18];
  const float* mlp_b3 = (const float*)d_in[19];
  float* out = (float*)d_out;

  // workspace layout (bytes); peak ~514 MB
  const size_t SZ_H    = (size_t)N_NODES * HID * 4;               //  51.2 MB
  const size_t SZ_CNT  = (size_t)N_NODES * NUM_REL * 4;           //   1.6 MB
  const size_t SZ_AGG1 = (size_t)N_NODES * NUM_REL * IN_DIM * 4;  // 204.8 MB
  const size_t SZ_AGG2 = (size_t)N_NODES * NUM_REL * HID * 4;     // 409.6 MB
  const size_t SZ_Z1   = (size_t)M_DEC * HID * 4;                 // 204.8 MB

  char* ws = (char*)d_ws;
  float* h1  = (float*)(ws);                       // [50000,256]
  float* h2  = (float*)(ws + SZ_H);                // [50000,256]
  float* cnt = (float*)(ws + 2 * SZ_H);            // [50000,8]
  float* agg = (float*)(ws + 2 * SZ_H + SZ_CNT);   // up to [50000,2048]
  float* z1  = agg;                                // reuse after layer GEMMs
  float* z2  = (float*)((char*)z1 + SZ_Z1);        // [200000,128]

  const dim3 blk256(256), blk128(128);

  // ======================= layer 1 =======================
  hipMemsetAsync(cnt, 0, SZ_CNT, stream);
  hipMemsetAsync(agg, 0, SZ_AGG1, stream);
  {
    long total = (long)NUM_EDGES * IN_DIM;
    rgcn_agg_kernel<<<dim3((total + 255) / 256), blk256, 0, stream>>>(
        x, ei, et, agg, cnt, NUM_EDGES, IN_DIM, 7);
    rcnt_kernel<<<dim3((N_NODES * NUM_REL + 255) / 256), blk256, 0, stream>>>(
        cnt, N_NODES * NUM_REL);
  }
  {
    dim3 grid((N_NODES + 63) / 64, HID / 128);
    // h1 = x @ root1 + b1
    gemm_wmma_kernel<0, 0, 1, 0><<<grid, blk128, 0, stream>>>(
        h1, x, root1, b1, nullptr, nullptr, N_NODES, HID, IN_DIM, 0);
    // h1 += (agg * rcnt) @ W1_flat
    gemm_wmma_kernel<1, 0, 0, 1><<<grid, blk128, 0, stream>>>(
        h1, agg, W1, nullptr, cnt, nullptr, N_NODES, HID, NUM_REL * IN_DIM, 7);
  }
  relu_ln_kernel<false><<<dim3((N_NODES + 7) / 8), blk256, 0, stream>>>(
      h1, ln1_g, ln1_b, h1, N_NODES);

  // ======================= layer 2 =======================
  hipMemsetAsync(cnt, 0, SZ_CNT, stream);
  hipMemsetAsync(agg, 0, SZ_AGG2, stream);
  {
    long total = (long)NUM_EDGES * HID;
    rgcn_agg_kernel<<<dim3((total + 255) / 256), blk256, 0, stream>>>(
        h1, ei, et, agg, cnt, NUM_EDGES, HID, 8);
    rcnt_kernel<<<dim3((N_NODES * NUM_REL + 255) / 256), blk256, 0, stream>>>(
        cnt, N_NODES * NUM_REL);
  }
  {
    dim3 grid((N_NODES + 63) / 64, HID / 128);
    gemm_wmma_kernel<0, 0, 1, 0><<<grid, blk128, 0, stream>>>(
        h2, h1, root2, b2, nullptr, nullptr, N_NODES, HID, HID, 0);
    gemm_wmma_kernel<1, 0, 0, 1><<<grid, blk128, 0, stream>>>(
        h2, agg, W2, nullptr, cnt, nullptr, N_NODES, HID, NUM_REL * HID, 8);
  }
  // h1 := h1 + LN(relu(h2))   (h = h1 + h2)
  relu_ln_kernel<true><<<dim3((N_NODES + 7) / 8), blk256, 0, stream>>>(
      h2, ln2_g, ln2_b, h1, N_NODES);

  // ======================= decode MLP =======================
  {
    // z1 = gelu(concat(h[dec0], h[dec1]) @ mlp_w1 + mlp_b1), K=512
    dim3 grid((M_DEC + 63) / 64, HID / 128);
    gemm_wmma_kernel<2, 1, 1, 0><<<grid, blk128, 0, stream>>>(
        z1, h1, mlp_w1, mlp_b1, nullptr, dec, M_DEC, HID, 2 * HID, 0);
  }
  {
    // z2 = gelu(z1 @ mlp_w2 + mlp_b2), N=128, K=256
    dim3 grid((M_DEC + 63) / 64, 128 / 128);
    gemm_wmma_kernel<0, 1, 1, 0><<<grid, blk128, 0, stream>>>(
        z2, z1, mlp_w2, mlp_b2, nullptr, nullptr, M_DEC, 128, HID, 0);
  }
  final_fc_kernel<<<dim3((M_DEC + 255) / 256), blk256, 0, stream>>>(
      z2, mlp_w3, mlp_b3, out, M_DEC);
}